// TensorGuidedFilter_61727269978413
// MI455X (gfx1250) — compile-verified
//
#include <hip/hip_runtime.h>
#include <hip/hip_bf16.h>
#include <stdint.h>

// ---------------- guided filter, fused single pass ----------------
// radius 8, eps 0.1, planes of 1024x1024 fp32.
#define RAD   8
#define KTAP  (2 * RAD + 1)          // 17
#define TLE   64                     // output tile edge
#define FULL  (TLE + 4 * RAD)        // 96  (input tile incl. double halo)
#define MID   (TLE + 2 * RAD)        // 80  (region where a,b are valid)
#define FS    (FULL + 1)             // 97  padded LDS stride
#define MS    (MID + 1)              // 81  padded LDS stride
#define NT    256
#define INVK  (1.0f / 17.0f)
#define GF_EPS 0.1f

#define AS1 __attribute__((address_space(1)))
#define AS3 __attribute__((address_space(3)))

#if defined(__HIP_DEVICE_COMPILE__) && defined(__gfx1250__) && \
    __has_builtin(__builtin_amdgcn_global_load_async_to_lds_b32)
#define USE_ASYNC_LDS 1
#endif

__device__ __forceinline__ int reflect_idx(int i, int n) {
  i = (i < 0) ? -i : i;                 // jnp reflect: -1 -> 1
  i = (i >= n) ? (2 * n - 2 - i) : i;   // n -> n-2
  return i;
}

// quantity selector for the first round of box filters
// mode 0: I   1: p   2: I*p   3: I*I
__device__ __forceinline__ float qval(const float* __restrict__ a,
                                      const float* __restrict__ b,
                                      int idx, int mode) {
  float x = a[idx];
  if (mode == 0) return x;
  if (mode == 3) return x * x;
  float y = b[idx];
  return (mode == 1) ? y : x * y;
}

__global__ __launch_bounds__(NT)
void guided_filter_kernel(const float* __restrict__ p_in,   // input_img
                          const float* __restrict__ I_in,   // guide_img
                          float* __restrict__ out,
                          int H, int W) {
  // ---- LDS plan (~215 KB of the 320 KB/WGP budget) ----
  __shared__ float sI [FULL * FS];   // guide tile   (96x96, stride 97)
  __shared__ float sP [FULL * FS];   // input tile
  __shared__ float mI [MID * MS];    // mean_I       (80x80, stride 81)
  __shared__ float mP [MID * MS];    // mean_p  -> later b
  __shared__ float mIp[MID * MS];    // mean_Ip -> later a
  __shared__ float mII[MID * MS];    // mean_II
  __shared__ float scr[2 * TLE * MS]; // 10368 floats: vertical-pass scratch,
                                      // then tA/tB for the second box round

  const int tid   = threadIdx.x;
  const int plane = blockIdx.z;
  const long planeOff = (long)plane * H * W;
  const int oy0 = blockIdx.y * TLE;
  const int ox0 = blockIdx.x * TLE;
  const int y0  = oy0 - 2 * RAD;
  const int x0  = ox0 - 2 * RAD;

  // ================= Stage A: async load 96x96 tiles of I and p =============
  for (int i = tid; i < FULL * FULL; i += NT) {
    int ly = i / FULL, lx = i - ly * FULL;
    int gy = reflect_idx(y0 + ly, H);
    int gx = reflect_idx(x0 + lx, W);
    long off = planeOff + (long)gy * W + gx;
    int d = ly * FS + lx;
#ifdef USE_ASYNC_LDS
    __builtin_amdgcn_global_load_async_to_lds_b32(
        (AS1 int*)(I_in + off), (AS3 int*)&sI[d], 0, 0);
    __builtin_amdgcn_global_load_async_to_lds_b32(
        (AS1 int*)(p_in + off), (AS3 int*)&sP[d], 0, 0);
#else
    sI[d] = I_in[off];
    sP[d] = p_in[off];
#endif
  }
#ifdef USE_ASYNC_LDS
#if __has_builtin(__builtin_amdgcn_s_wait_asynccnt)
  __builtin_amdgcn_s_wait_asynccnt(0);
#else
  asm volatile("s_wait_asynccnt 0" ::: "memory");
#endif
#endif
  __syncthreads();

  // ====== Stage B: box(I), box(p), box(I*p), box(I*I) via sliding sums ======
  for (int mode = 0; mode < 4; ++mode) {
    float* dst = (mode == 0) ? mI : (mode == 1) ? mP : (mode == 2) ? mIp : mII;

    // vertical: 96 cols x 80 out rows, split rows into 2 chunks -> 192 workers
    for (int w = tid; w < FULL * 2; w += NT) {
      int c  = w % FULL;
      int yb = (w / FULL) * (MID / 2);
      int ye = yb + (MID / 2);
      float s = 0.f;
#pragma unroll
      for (int j = 0; j < KTAP; ++j) s += qval(sI, sP, (yb + j) * FS + c, mode);
      scr[yb * FS + c] = s * INVK;
      for (int y = yb + 1; y < ye; ++y) {
        s += qval(sI, sP, (y + KTAP - 1) * FS + c, mode)
           - qval(sI, sP, (y - 1) * FS + c, mode);
        scr[y * FS + c] = s * INVK;
      }
    }
    __syncthreads();

    // horizontal: 80 rows x 80 out cols, 2 col chunks -> 160 workers
    for (int w = tid; w < MID * 2; w += NT) {
      int r  = w % MID;
      int xb = (w / MID) * (MID / 2);
      int xe = xb + (MID / 2);
      const float* row = &scr[r * FS];
      float s = 0.f;
#pragma unroll
      for (int j = 0; j < KTAP; ++j) s += row[xb + j];
      dst[r * MS + xb] = s * INVK;
      for (int x = xb + 1; x < xe; ++x) {
        s += row[x + KTAP - 1] - row[x - 1];
        dst[r * MS + x] = s * INVK;
      }
    }
    __syncthreads();
  }

  // ================= Stage C: a = cov/(var+eps), b = mp - a*mi ==============
  for (int i = tid; i < MID * MID; i += NT) {
    int r = i / MID, c = i - r * MID;
    int idx = r * MS + c;
    float meanI = mI[idx], meanP = mP[idx];
    float cov = mIp[idx] - meanI * meanP;
    float var = mII[idx] - meanI * meanI;
    float a = cov / (var + GF_EPS);
    float b = meanP - a * meanI;
    mIp[idx] = a;   // reuse buffer as 'a'
    mP[idx]  = b;   // reuse buffer as 'b'
  }
  __syncthreads();

  // ================= Stage D: box(a), box(b), fuse output ===================
  float* tA = scr;
  float* tB = scr + TLE * MS;

  // vertical: 80 cols x 64 out rows, 2 row chunks -> 160 workers
  for (int w = tid; w < MID * 2; w += NT) {
    int c  = w % MID;
    int yb = (w / MID) * (TLE / 2);
    int ye = yb + (TLE / 2);
    float sa = 0.f, sb = 0.f;
#pragma unroll
    for (int j = 0; j < KTAP; ++j) {
      sa += mIp[(yb + j) * MS + c];
      sb += mP [(yb + j) * MS + c];
    }
    tA[yb * MS + c] = sa * INVK;
    tB[yb * MS + c] = sb * INVK;
    for (int y = yb + 1; y < ye; ++y) {
      sa += mIp[(y + KTAP - 1) * MS + c] - mIp[(y - 1) * MS + c];
      sb += mP [(y + KTAP - 1) * MS + c] - mP [(y - 1) * MS + c];
      tA[y * MS + c] = sa * INVK;
      tB[y * MS + c] = sb * INVK;
    }
  }
  __syncthreads();

  // horizontal + fuse: out = mean_a * I + mean_b, into fin (reuse mI)
  float* fin = mI;   // 64x64, stride 64
  for (int w = tid; w < TLE * 2; w += NT) {
    int r  = w % TLE;
    int xb = (w / TLE) * (TLE / 2);
    int xe = xb + (TLE / 2);
    const float* ra = &tA[r * MS];
    const float* rb = &tB[r * MS];
    const float* Irow = &sI[(r + 2 * RAD) * FS + 2 * RAD];
    float sa = 0.f, sb = 0.f;
#pragma unroll
    for (int j = 0; j < KTAP; ++j) { sa += ra[xb + j]; sb += rb[xb + j]; }
    fin[r * TLE + xb] = (sa * INVK) * Irow[xb] + (sb * INVK);
    for (int x = xb + 1; x < xe; ++x) {
      sa += ra[x + KTAP - 1] - ra[x - 1];
      sb += rb[x + KTAP - 1] - rb[x - 1];
      fin[r * TLE + x] = (sa * INVK) * Irow[x] + (sb * INVK);
    }
  }
  __syncthreads();

  // ================= coalesced store ========================================
  for (int i = tid; i < TLE * TLE; i += NT) {
    int r = i / TLE, c = i - r * TLE;
    int gy = oy0 + r, gx = ox0 + c;
    if (gy < H && gx < W) out[planeOff + (long)gy * W + gx] = fin[i];
  }
}

extern "C" void kernel_launch(void* const* d_in, const int* in_sizes, int n_in,
                              void* d_out, int out_size, void* d_ws, size_t ws_size,
                              hipStream_t stream) {
  (void)n_in; (void)d_ws; (void)ws_size; (void)out_size;
  const float* p_img = (const float*)d_in[0];   // input_img
  const float* I_img = (const float*)d_in[1];   // guide_img
  float* out = (float*)d_out;

  const int H = 1024, W = 1024;
  const int planes = in_sizes[0] / (H * W);     // B*C = 24

  dim3 grid((W + TLE - 1) / TLE, (H + TLE - 1) / TLE, planes);
  guided_filter_kernel<<<grid, NT, 0, stream>>>(p_img, I_img, out, H, W);
}